// PolymerGCN_44057774522484
// MI455X (gfx1250) — compile-verified
//
#include <hip/hip_runtime.h>

#define H_DIM 128

typedef float v2f __attribute__((ext_vector_type(2)));
typedef float v8f __attribute__((ext_vector_type(8)));

// ---------------------------------------------------------------------------
// GEMM: C[M x 128] = X[M x K] @ W[K x 128], fp32 WMMA 16x16x4, one wave/tile.
// M must be a multiple of 16 (N=100000 = 6250*16). EXEC is all-ones.
// ---------------------------------------------------------------------------
template <int K>
__global__ __launch_bounds__(256) void gemm_wmma_f32(
    const float* __restrict__ X, const float* __restrict__ W,
    float* __restrict__ C, int M) {
  const int TN = H_DIM / 16;  // 8 column tiles
  long long gtid = (long long)blockIdx.x * blockDim.x + threadIdx.x;
  int wave = (int)(gtid >> 5);
  int lane = threadIdx.x & 31;
  int tileM = wave / TN;
  int tileN = wave - tileM * TN;
  if (tileM * 16 >= M) return;  // wave-uniform

  int half = lane >> 4;   // 0: K pair {kk,kk+1}, 1: {kk+2,kk+3}
  int l15 = lane & 15;
  int rowA = tileM * 16 + l15;
  int colB = tileN * 16 + l15;
  const float* xrow = X + (size_t)rowA * K;

  v8f c = {0.f, 0.f, 0.f, 0.f, 0.f, 0.f, 0.f, 0.f};
#pragma unroll
  for (int kk = 0; kk < K; kk += 4) {
    int ka = kk + 2 * half;
    v2f a = {xrow[ka], xrow[ka + 1]};
    v2f b = {W[(size_t)ka * H_DIM + colB], W[(size_t)(ka + 1) * H_DIM + colB]};
    c = __builtin_amdgcn_wmma_f32_16x16x4_f32(false, a, false, b, (short)0, c,
                                              false, false);
  }
  // C/D layout: VGPR r -> lanes 0-15: M=r, lanes 16-31: M=r+8
  int outRow = tileM * 16 + half * 8;
#pragma unroll
  for (int r = 0; r < 8; ++r)
    C[(size_t)(outRow + r) * H_DIM + colB] = c[r];
}

// ---------------------------------------------------------------------------
// Degree / normalization precompute
// ---------------------------------------------------------------------------
__global__ void fill1_k(float* __restrict__ p, int n) {
  int i = blockIdx.x * blockDim.x + threadIdx.x;
  if (i < n) p[i] = 1.0f;  // self-loop contributes 1
}

__global__ void deg_count_k(const long long* __restrict__ dst,
                            float* __restrict__ deg, int e) {
  int i = blockIdx.x * blockDim.x + threadIdx.x;
  if (i < e) atomicAdd(&deg[dst[i]], 1.0f);
}

__global__ void rsqrt_inplace_k(float* __restrict__ p, int n) {
  int i = blockIdx.x * blockDim.x + threadIdx.x;
  if (i < n) p[i] = rsqrtf(fmaxf(p[i], 1.0f));
}

__global__ void edge_norm_k(const long long* __restrict__ src,
                            const long long* __restrict__ dst,
                            const float* __restrict__ dinv,
                            float* __restrict__ enorm, int e) {
  int i = blockIdx.x * blockDim.x + threadIdx.x;
  if (i < e) enorm[i] = dinv[src[i]] * dinv[dst[i]];
}

// ---------------------------------------------------------------------------
// Aggregation: agg = dinv^2 * h (self loop) + bias, then edge scatter-adds.
// ---------------------------------------------------------------------------
__global__ void selfloop_bias_k(const float* __restrict__ h,
                                const float* __restrict__ dinv,
                                const float* __restrict__ b,
                                float* __restrict__ agg, long long total) {
  long long idx = (long long)blockIdx.x * blockDim.x + threadIdx.x;
  if (idx >= total) return;
  int i = (int)(idx >> 7);
  int f = (int)(idx & 127);
  float d = dinv[i];
  agg[idx] = h[idx] * d * d + b[f];
}

// one wave32 per edge; each lane moves 4 features (float4 = coalesced 512B/edge)
__global__ void edge_scatter_k(const long long* __restrict__ src,
                               const long long* __restrict__ dst,
                               const float* __restrict__ enorm,
                               const float* __restrict__ h,
                               float* __restrict__ agg, int e) {
  long long gtid = (long long)blockIdx.x * blockDim.x + threadIdx.x;
  int wave = (int)(gtid >> 5);
  int lane = threadIdx.x & 31;
  if (wave >= e) return;
  long long s = src[wave];
  long long d = dst[wave];
  float nm = enorm[wave];
  const float4* hs = (const float4*)(h + (size_t)s * H_DIM);
  float4 v = hs[lane];
  float* ad = agg + (size_t)d * H_DIM + lane * 4;
  atomicAdd(ad + 0, v.x * nm);
  atomicAdd(ad + 1, v.y * nm);
  atomicAdd(ad + 2, v.z * nm);
  atomicAdd(ad + 3, v.w * nm);
}

// ---------------------------------------------------------------------------
// BatchNorm (training-mode batch stats) + ReLU
// ---------------------------------------------------------------------------
__global__ void bn_zero_k(float* __restrict__ s, float* __restrict__ q) {
  int f = threadIdx.x;
  s[f] = 0.0f;
  q[f] = 0.0f;
}

__global__ void colstats_k(const float* __restrict__ agg,
                           float* __restrict__ sum, float* __restrict__ sumsq,
                           int n) {
  int f = threadIdx.x;  // blockDim.x == 128
  float s = 0.f, q = 0.f;
  for (int r = blockIdx.x; r < n; r += gridDim.x) {
    float v = agg[(size_t)r * H_DIM + f];
    s += v;
    q += v * v;
  }
  atomicAdd(&sum[f], s);
  atomicAdd(&sumsq[f], q);
}

__global__ void bn_finalize_k(const float* __restrict__ sum,
                              const float* __restrict__ sumsq,
                              const float* __restrict__ gamma,
                              const float* __restrict__ beta,
                              float* __restrict__ scale,
                              float* __restrict__ shift, float invN) {
  int f = threadIdx.x;
  float m = sum[f] * invN;
  float var = fmaxf(sumsq[f] * invN - m * m, 0.0f);
  float sc = gamma[f] * rsqrtf(var + 1e-5f);
  scale[f] = sc;
  shift[f] = beta[f] - m * sc;
}

__global__ void bn_relu_k(const float* __restrict__ agg,
                          const float* __restrict__ scale,
                          const float* __restrict__ shift,
                          float* __restrict__ xout, long long total) {
  long long idx = (long long)blockIdx.x * blockDim.x + threadIdx.x;
  if (idx >= total) return;
  int f = (int)(idx & 127);
  xout[idx] = fmaxf(agg[idx] * scale[f] + shift[f], 0.0f);
}

// ---------------------------------------------------------------------------
// Global mean pool + MLP head
// ---------------------------------------------------------------------------
__global__ void pool_acc_k(const float* __restrict__ x,
                           const long long* __restrict__ batch,
                           float* __restrict__ pooled,
                           float* __restrict__ counts, int n) {
  long long gtid = (long long)blockIdx.x * blockDim.x + threadIdx.x;
  int node = (int)(gtid >> 5);
  int lane = threadIdx.x & 31;
  if (node >= n) return;
  long long g = batch[node];
  const float4* xs = (const float4*)(x + (size_t)node * H_DIM);
  float4 v = xs[lane];
  float* pd = pooled + (size_t)g * H_DIM + lane * 4;
  atomicAdd(pd + 0, v.x);
  atomicAdd(pd + 1, v.y);
  atomicAdd(pd + 2, v.z);
  atomicAdd(pd + 3, v.w);
  if (lane == 0) atomicAdd(&counts[g], 1.0f);
}

__global__ void pool_div_k(float* __restrict__ pooled,
                           const float* __restrict__ counts, int g) {
  int idx = blockIdx.x * blockDim.x + threadIdx.x;
  if (idx >= g * H_DIM) return;
  pooled[idx] /= fmaxf(counts[idx >> 7], 1.0f);
}

__global__ void mlp1_k(const float* __restrict__ pooled,
                       const float* __restrict__ Wh1,
                       const float* __restrict__ bh1, float* __restrict__ hid,
                       int g) {
  int idx = blockIdx.x * blockDim.x + threadIdx.x;  // g * 64 threads
  if (idx >= g * 64) return;
  int row = idx >> 6;
  int j = idx & 63;
  float acc = bh1[j];
  const float* pr = pooled + (size_t)row * H_DIM;
#pragma unroll 8
  for (int k = 0; k < H_DIM; ++k) acc += pr[k] * Wh1[k * 64 + j];
  hid[idx] = fmaxf(acc, 0.0f);
}

__global__ void mlp2_k(const float* __restrict__ hid,
                       const float* __restrict__ Wh2,
                       const float* __restrict__ bh2, float* __restrict__ out,
                       int g) {
  int row = blockIdx.x * blockDim.x + threadIdx.x;
  if (row >= g) return;
  float acc = bh2[0];
  const float* hr = hid + (size_t)row * 64;
#pragma unroll 8
  for (int k = 0; k < 64; ++k) acc += hr[k] * Wh2[k];
  out[row] = acc;
}

// ---------------------------------------------------------------------------
// Host-side orchestration
// ---------------------------------------------------------------------------
extern "C" void kernel_launch(void* const* d_in, const int* in_sizes, int n_in,
                              void* d_out, int out_size, void* d_ws,
                              size_t ws_size, hipStream_t stream) {
  const float* x_in = (const float*)d_in[0];
  const long long* ei = (const long long*)d_in[1];   // int64 [2,E]
  const long long* batch = (const long long*)d_in[2];  // int64 [N]
  const float* W[3] = {(const float*)d_in[3], (const float*)d_in[7],
                       (const float*)d_in[11]};
  const float* bias[3] = {(const float*)d_in[4], (const float*)d_in[8],
                          (const float*)d_in[12]};
  const float* gam[3] = {(const float*)d_in[5], (const float*)d_in[9],
                         (const float*)d_in[13]};
  const float* bet[3] = {(const float*)d_in[6], (const float*)d_in[10],
                         (const float*)d_in[14]};
  const float* Wh1 = (const float*)d_in[15];
  const float* bh1 = (const float*)d_in[16];
  const float* Wh2 = (const float*)d_in[17];
  const float* bh2 = (const float*)d_in[18];

  const int N = in_sizes[2];
  const int E = in_sizes[1] / 2;
  const int G = out_size;
  const long long* src = ei;
  const long long* dst = ei + E;

  // workspace layout (floats)
  float* ws = (float*)d_ws;
  size_t off = 0;
  float* hbuf = ws + off;   off += (size_t)N * H_DIM;
  float* agg = ws + off;    off += (size_t)N * H_DIM;
  float* xcur = ws + off;   off += (size_t)N * H_DIM;
  float* dinv = ws + off;   off += (size_t)N;
  float* enorm = ws + off;  off += (size_t)E;
  float* bnsum = ws + off;  off += H_DIM;
  float* bnsq = ws + off;   off += H_DIM;
  float* bnscale = ws + off; off += H_DIM;
  float* bnshift = ws + off; off += H_DIM;
  float* pooled = ws + off; off += (size_t)G * H_DIM;
  float* counts = ws + off; off += (size_t)G;  // contiguous after pooled
  float* hid = ws + off;    off += (size_t)G * 64;
  (void)ws_size; (void)n_in;

  const int T = 256;

  // degree -> dinv -> per-edge norm (shared by all 3 layers)
  fill1_k<<<(N + T - 1) / T, T, 0, stream>>>(dinv, N);
  deg_count_k<<<(E + T - 1) / T, T, 0, stream>>>(dst, dinv, E);
  rsqrt_inplace_k<<<(N + T - 1) / T, T, 0, stream>>>(dinv, N);
  edge_norm_k<<<(E + T - 1) / T, T, 0, stream>>>(src, dst, dinv, enorm, E);

  const long long tot = (long long)N * H_DIM;
  const int tiles = (N / 16) * (H_DIM / 16);
  const int gemm_blocks = (int)(((long long)tiles * 32 + T - 1) / T);
  const int scat_blocks = (int)(((long long)E * 32 + T - 1) / T);
  const int elem_blocks = (int)((tot + T - 1) / T);

  for (int l = 0; l < 3; ++l) {
    if (l == 0)
      gemm_wmma_f32<32><<<gemm_blocks, T, 0, stream>>>(x_in, W[0], hbuf, N);
    else
      gemm_wmma_f32<128><<<gemm_blocks, T, 0, stream>>>(xcur, W[l], hbuf, N);

    selfloop_bias_k<<<elem_blocks, T, 0, stream>>>(hbuf, dinv, bias[l], agg, tot);
    edge_scatter_k<<<scat_blocks, T, 0, stream>>>(src, dst, enorm, hbuf, agg, E);

    bn_zero_k<<<1, H_DIM, 0, stream>>>(bnsum, bnsq);
    colstats_k<<<1024, H_DIM, 0, stream>>>(agg, bnsum, bnsq, N);
    bn_finalize_k<<<1, H_DIM, 0, stream>>>(bnsum, bnsq, gam[l], bet[l],
                                           bnscale, bnshift, 1.0f / (float)N);
    bn_relu_k<<<elem_blocks, T, 0, stream>>>(agg, bnscale, bnshift, xcur, tot);
  }

  // zero pooled sums + counts (contiguous), then segment-mean pool
  hipMemsetAsync(pooled, 0, ((size_t)G * H_DIM + G) * sizeof(float), stream);
  pool_acc_k<<<(int)(((long long)N * 32 + T - 1) / T), T, 0, stream>>>(
      xcur, batch, pooled, counts, N);
  pool_div_k<<<(G * H_DIM + T - 1) / T, T, 0, stream>>>(pooled, counts, G);

  mlp1_k<<<(G * 64 + T - 1) / T, T, 0, stream>>>(pooled, Wh1, bh1, hid, G);
  mlp2_k<<<(G + T - 1) / T, T, 0, stream>>>(hid, Wh2, bh2, (float*)d_out, G);
}